// TopKGate_83305185673388
// MI455X (gfx1250) — compile-verified
//
#include <hip/hip_runtime.h>
#include <hip/hip_bf16.h>

// ---------------------------------------------------------------------------
// Top-1 MoE gate for MI455X (gfx1250, wave32).
//   logits[T,E] = x[T,D] @ wg[E,D]^T      (fp32, via V_WMMA_F32_16X16X4_F32)
//   idx[t]  = argmax_e logits[t,e]        (first-max tie rule)
//   gate[t] = softmax(logits[t,:])[idx[t]] = 1 / sum_e exp(l_e - l_max)
//
// Memory-bound: x (134 MB fp32) is read exactly once from HBM (~5.8 us at
// 23.3 TB/s); wg (1 MB) lives in WGP$/L2. FP32 WMMA keeps exact fp32
// accumulation (no precision drop vs the fp32 reference).
//
// Register blocking: each wave owns TWO 16-token tiles (32 tokens) and all
// 64 experts -> per K-step: 2 A loads + 4 B loads feed 8 WMMAs (0.75
// loads/WMMA), with B fragments reused from registers across both M tiles.
// Epilogue uses all 32 lanes: one token per lane.
// ---------------------------------------------------------------------------

typedef __attribute__((ext_vector_type(2))) float v2f;
typedef __attribute__((ext_vector_type(8))) float v8f;

#define DMODEL 4096
#define NEXP 64
#define WAVES_PER_BLOCK 4
#define TOKENS_PER_WAVE 32
#define TOKENS_PER_BLOCK (WAVES_PER_BLOCK * TOKENS_PER_WAVE)

__global__ __launch_bounds__(WAVES_PER_BLOCK * 32, 2)
void topk_gate_wmma_kernel(const float* __restrict__ x,
                           const float* __restrict__ wg,
                           float* __restrict__ out,   // [0,T): idx as float, [T,2T): gate
                           int T) {
    __shared__ float lds[WAVES_PER_BLOCK * TOKENS_PER_WAVE * NEXP]; // 32 KB

    const int lane = threadIdx.x & 31;
    const int wave = threadIdx.x >> 5;
    const int h    = lane >> 4;   // half-wave: selects K pair (A/B) and M group (C/D)
    const int l16  = lane & 15;

    const int rowBase = blockIdx.x * TOKENS_PER_BLOCK + wave * TOKENS_PER_WAVE;

    // Clamp rows so partial tiles stay in-bounds (WMMA needs EXEC all-ones,
    // so we cannot mask lanes; duplicate the last row instead).
    int rowA0 = rowBase + l16;
    int rowA1 = rowBase + 16 + l16;
    if (rowA0 > T - 1) rowA0 = T - 1;
    if (rowA1 > T - 1) rowA1 = T - 1;

    // A-fragments (16x4 f32): lane<16 holds K={k,k+1}, lane>=16 holds K={k+2,k+3}
    const float* a0Ptr = x + (size_t)rowA0 * DMODEL + 2 * h;
    const float* a1Ptr = x + (size_t)rowA1 * DMODEL + 2 * h;
    // B-fragments (4x16 f32): expert row = tile*16 + (lane&15), same K split
    const float* b0 = wg + (size_t)(l16 +  0) * DMODEL + 2 * h;
    const float* b1 = wg + (size_t)(l16 + 16) * DMODEL + 2 * h;
    const float* b2 = wg + (size_t)(l16 + 32) * DMODEL + 2 * h;
    const float* b3 = wg + (size_t)(l16 + 48) * DMODEL + 2 * h;

    // Accumulators: [m-tile][n-tile], 8 x v8f = 64 VGPRs.
    v8f accA0 = {}, accA1 = {}, accA2 = {}, accA3 = {};
    v8f accB0 = {}, accB1 = {}, accB2 = {}, accB3 = {};

    #pragma unroll 4
    for (int k = 0; k < DMODEL; k += 4) {
        v2f a0  = *(const v2f*)(a0Ptr + k);
        v2f a1  = *(const v2f*)(a1Ptr + k);
        v2f bb0 = *(const v2f*)(b0 + k);
        v2f bb1 = *(const v2f*)(b1 + k);
        v2f bb2 = *(const v2f*)(b2 + k);
        v2f bb3 = *(const v2f*)(b3 + k);
        accA0 = __builtin_amdgcn_wmma_f32_16x16x4_f32(false, a0, false, bb0,
                                                      (short)0, accA0, false, false);
        accA1 = __builtin_amdgcn_wmma_f32_16x16x4_f32(false, a0, false, bb1,
                                                      (short)0, accA1, false, false);
        accA2 = __builtin_amdgcn_wmma_f32_16x16x4_f32(false, a0, false, bb2,
                                                      (short)0, accA2, false, false);
        accA3 = __builtin_amdgcn_wmma_f32_16x16x4_f32(false, a0, false, bb3,
                                                      (short)0, accA3, false, false);
        accB0 = __builtin_amdgcn_wmma_f32_16x16x4_f32(false, a1, false, bb0,
                                                      (short)0, accB0, false, false);
        accB1 = __builtin_amdgcn_wmma_f32_16x16x4_f32(false, a1, false, bb1,
                                                      (short)0, accB1, false, false);
        accB2 = __builtin_amdgcn_wmma_f32_16x16x4_f32(false, a1, false, bb2,
                                                      (short)0, accB2, false, false);
        accB3 = __builtin_amdgcn_wmma_f32_16x16x4_f32(false, a1, false, bb3,
                                                      (short)0, accB3, false, false);
    }

    // Spill 32x64 logits to LDS. C/D layout: VGPR r, lane -> M = r + 8*h,
    // N = (lane&15) within each 16-expert tile.
    float* myLds = lds + wave * (TOKENS_PER_WAVE * NEXP);
    #pragma unroll
    for (int r = 0; r < 8; ++r) {
        const int m = r + 8 * h;
        myLds[m * NEXP +  0 + l16] = accA0[r];
        myLds[m * NEXP + 16 + l16] = accA1[r];
        myLds[m * NEXP + 32 + l16] = accA2[r];
        myLds[m * NEXP + 48 + l16] = accA3[r];
        myLds[(m + 16) * NEXP +  0 + l16] = accB0[r];
        myLds[(m + 16) * NEXP + 16 + l16] = accB1[r];
        myLds[(m + 16) * NEXP + 32 + l16] = accB2[r];
        myLds[(m + 16) * NEXP + 48 + l16] = accB3[r];
    }

    __syncthreads();

    // Epilogue: each of the 32 lanes owns one token of the wave's 32-token tile.
    {
        const int token = rowBase + lane;
        if (token < T) {
            const float* rowL = myLds + lane * NEXP;
            float m = rowL[0];
            int   idx = 0;
            #pragma unroll 8
            for (int e = 1; e < NEXP; ++e) {
                const float v = rowL[e];
                if (v > m) { m = v; idx = e; }   // strict > keeps first max (JAX argmax)
            }
            float s = 0.0f;
            #pragma unroll 8
            for (int e = 0; e < NEXP; ++e) {
                s += __expf(rowL[e] - m);
            }
            out[token]     = (float)idx;   // indices1_s
            out[T + token] = 1.0f / s;     // gates1_s = softmax max = 1/sumexp
        }
    }
}

extern "C" void kernel_launch(void* const* d_in, const int* in_sizes, int n_in,
                              void* d_out, int out_size, void* d_ws, size_t ws_size,
                              hipStream_t stream) {
    const float* x  = (const float*)d_in[0];   // [T, 4096] fp32
    const float* wg = (const float*)d_in[1];   // [64, 4096] fp32
    float* out = (float*)d_out;                // [2*T] fp32 (idx, gate)

    const int T = in_sizes[0] / DMODEL;        // 8192 for the reference shapes
    const int grid = (T + TOKENS_PER_BLOCK - 1) / TOKENS_PER_BLOCK;

    topk_gate_wmma_kernel<<<grid, WAVES_PER_BLOCK * 32, 0, stream>>>(x, wg, out, T);
}